// RPN_64072322122233
// MI455X (gfx1250) — compile-verified
//
#include <hip/hip_runtime.h>
#include <hip/hip_bf16.h>

typedef unsigned long long ull;
typedef float v2f __attribute__((ext_vector_type(2)));
typedef float v8f __attribute__((ext_vector_type(8)));

// ---------------- problem constants ----------------
static constexpr int B     = 8;
static constexpr int N3    = 128*128*3;   // 49152
static constexpr int N4    = 64*64*3;     // 12288
static constexpr int N5    = 32*32*3;     // 3072
static constexpr int A     = N3+N4+N5;    // 64512
static constexpr int P3    = 65536, P4 = 16384, P5 = 4096;
static constexpr int NCAND = 3000;        // 3 levels x PRE_NMS(1000)
static constexpr int CPAD  = 4096;
static constexpr int POSTN = 1000;
static constexpr int M_GT  = 40;
#define SCALE_CLAMP 3.3322045101752038f   // log(224/8)
#define IMG_SZ 1024.0f

// ---------------- anchors (analytic) ----------------
__device__ __forceinline__ void anchor_box(int i, float& x0, float& y0, float& x1, float& y1) {
    int stride, n, li;
    if (i < N3)            { stride = 8;  n = 128; li = i; }
    else if (i < N3 + N4)  { stride = 16; n = 64;  li = i - N3; }
    else                   { stride = 32; n = 32;  li = i - N3 - N4; }
    int loc = li / 3, ar = li - loc * 3;
    float s4 = (float)stride * 4.0f;
    float cx = ((float)(loc % n) + 0.5f) * (float)stride;
    float cy = ((float)(loc / n) + 0.5f) * (float)stride;
    const float SQ2 = 1.41421356237309515f, ISQ2 = 0.70710678118654752f;
    float w = (ar == 0) ? s4 * SQ2 : (ar == 1 ? s4 : s4 * ISQ2);
    float h = (ar == 0) ? s4 * ISQ2 : (ar == 1 ? s4 : s4 * SQ2);
    x0 = cx - w * 0.5f; y0 = cy - h * 0.5f;
    x1 = cx + w * 0.5f; y1 = cy + h * 0.5f;
}

__device__ __forceinline__ float clamp01k(float v) {
    return fminf(fmaxf(v, 0.0f), IMG_SZ);
}

// ---------------- sort key packing ----------------
// 64-bit key, sorted DESCENDING: primary score desc, secondary index asc
// (matches jax top_k / stable argsort(-s) tie-breaking).
__device__ __forceinline__ ull make_pack(float s, unsigned int idx) {
    unsigned int u = __float_as_uint(s);
    unsigned int e = (u & 0x80000000u) ? ~u : (u | 0x80000000u); // monotone encode
    return ((ull)e << 32) | (ull)(0xFFFFFFFFu - idx);
}

// ---------------- kernels ----------------
__global__ void decode_kernel(const float* __restrict__ d3,
                              const float* __restrict__ d4,
                              const float* __restrict__ d5,
                              float4* __restrict__ dec) {
    int idx = blockIdx.x * blockDim.x + threadIdx.x;
    if (idx >= B * A) return;
    int b = idx / A, i = idx - b * A;
    const float* dp;
    if (i < N3)           dp = d3 + ((size_t)b * N3 + i) * 4;
    else if (i < N3 + N4) dp = d4 + ((size_t)b * N4 + (i - N3)) * 4;
    else                  dp = d5 + ((size_t)b * N5 + (i - N3 - N4)) * 4;
    float x0, y0, x1, y1;
    anchor_box(i, x0, y0, x1, y1);
    float dx = dp[0], dy = dp[1];
    float dw = fminf(dp[2], SCALE_CLAMP), dh = fminf(dp[3], SCALE_CLAMP);
    float px = (x0 + x1) * 0.5f, py = (y0 + y1) * 0.5f;
    float pw = x1 - x0, ph = y1 - y0;
    float gx = px + pw * dx, gy = py + ph * dy;
    float gw = pw * __expf(dw), gh = ph * __expf(dh);
    float4 o;
    o.x = clamp01k(gx - gw * 0.5f);
    o.y = clamp01k(gy - gh * 0.5f);
    o.z = clamp01k(gx + gw * 0.5f);
    o.w = clamp01k(gy + gh * 0.5f);
    dec[idx] = o;
}

__global__ void fill_pack_kernel(const float* __restrict__ scores,
                                 ull* __restrict__ pack, int n, int P) {
    int t = blockIdx.x * blockDim.x + threadIdx.x;
    if (t >= B * P) return;
    int b = t / P, i = t - b * P;
    pack[t] = (i < n) ? make_pack(scores[(size_t)b * n + i], (unsigned)i) : 0ULL;
}

// one bitonic compare-exchange pass over global memory (j > 1024)
__global__ void bitonic_global(ull* __restrict__ a, int total, int segMask, int k, int j) {
    int t = blockIdx.x * blockDim.x + threadIdx.x;
    if (t >= total) return;
    int p = t ^ j;
    if (p > t) {
        int s = t & segMask;
        ull x = a[t], y = a[p];
        bool desc = ((s & k) == 0);
        if (desc ? (x < y) : (x > y)) { a[t] = y; a[p] = x; }
    }
}

// fused passes j = jstart .. 1 inside LDS (2048 elements / block)
__global__ void bitonic_fused(ull* __restrict__ a, int segMask, int k, int jstart) {
    __shared__ ull lds[2048];
    int base = blockIdx.x * 2048;
    lds[threadIdx.x]        = a[base + threadIdx.x];
    lds[threadIdx.x + 1024] = a[base + threadIdx.x + 1024];
    for (int j = jstart; j >= 1; j >>= 1) {
        __syncthreads();
        int h  = threadIdx.x;
        int i1 = ((h & ~(j - 1)) << 1) | (h & (j - 1));
        int i2 = i1 + j;
        int s  = (base + i1) & segMask;
        ull x = lds[i1], y = lds[i2];
        bool desc = ((s & k) == 0);
        if (desc ? (x < y) : (x > y)) { lds[i1] = y; lds[i2] = x; }
    }
    __syncthreads();
    a[base + threadIdx.x]        = lds[threadIdx.x];
    a[base + threadIdx.x + 1024] = lds[threadIdx.x + 1024];
}

// build the 3000 candidates/image (top-1000 per level, concat order)
__global__ void gather_cand(const ull* __restrict__ p3, const ull* __restrict__ p4,
                            const ull* __restrict__ p5, const float4* __restrict__ dec,
                            ull* __restrict__ candPack, float4* __restrict__ candBox) {
    int t = blockIdx.x * blockDim.x + threadIdx.x;
    if (t >= B * CPAD) return;
    int b = t / CPAD, s = t - b * CPAD;
    if (s >= NCAND) { candPack[t] = 0ULL; return; }
    int level = s / 1000, rank = s - level * 1000;
    ull p; int off;
    if (level == 0)      { p = p3[(size_t)b * P3 + rank]; off = 0; }
    else if (level == 1) { p = p4[(size_t)b * P4 + rank]; off = N3; }
    else                 { p = p5[(size_t)b * P5 + rank]; off = N3 + N4; }
    unsigned idx = 0xFFFFFFFFu - (unsigned)(p & 0xFFFFFFFFull);
    candBox[(size_t)b * NCAND + s] = dec[(size_t)b * A + off + idx];
    // keep score encoding, new secondary key = concat position (stable argsort)
    candPack[t] = (p & 0xFFFFFFFF00000000ull) | (ull)(0xFFFFFFFFu - (unsigned)s);
}

__global__ void gather_sorted(const ull* __restrict__ candPack,
                              const float4* __restrict__ candBox,
                              float4* __restrict__ sortedBox) {
    int t = blockIdx.x * blockDim.x + threadIdx.x;
    if (t >= B * NCAND) return;
    int b = t / NCAND, r = t - b * NCAND;
    ull p = candPack[(size_t)b * CPAD + r];
    unsigned pos = 0xFFFFFFFFu - (unsigned)(p & 0xFFFFFFFFull);
    sortedBox[t] = candBox[(size_t)b * NCAND + pos];
}

// greedy NMS, one block per image; writes proposals to d_out.
// IoU test is division-free: inter/union > t  <=>  inter > t*union (union>0;
// degenerate 0/0 compares false either way).
__global__ void __launch_bounds__(1024)
nms_kernel(const float4* __restrict__ sortedBox, float* __restrict__ outProps) {
    __shared__ float4 boxes[NCAND];
    __shared__ unsigned char supp[NCAND];
    __shared__ int keeplist[POSTN];
    __shared__ int nkeep;
    int b = blockIdx.x;
    for (int i = threadIdx.x; i < NCAND; i += blockDim.x) {
        boxes[i] = sortedBox[(size_t)b * NCAND + i];
        supp[i] = 0;
    }
    if (threadIdx.x == 0) nkeep = 0;
    __syncthreads();
    for (int i = 0; i < NCAND; i++) {
        bool alive = (supp[i] == 0);   // uniform: read from LDS after barrier
        if (alive) {
            if (threadIdx.x == 0 && nkeep < POSTN) keeplist[nkeep++] = i;
            float4 bi = boxes[i];
            float ai = (bi.z - bi.x) * (bi.w - bi.y);
            for (int jj = i + 1 + threadIdx.x; jj < NCAND; jj += blockDim.x) {
                float4 bj = boxes[jj];
                float iw = fmaxf(fminf(bi.z, bj.z) - fmaxf(bi.x, bj.x), 0.0f);
                float ih = fmaxf(fminf(bi.w, bj.w) - fmaxf(bi.y, bj.y), 0.0f);
                float inter = iw * ih;
                float aj = (bj.z - bj.x) * (bj.w - bj.y);
                // inter/(ai+aj-inter) > 0.7  (division-free)
                if (inter > 0.7f * (ai + aj - inter)) supp[jj] = 1;
            }
        }
        __syncthreads();
    }
    int nk = nkeep;
    for (int kk = threadIdx.x; kk < POSTN; kk += blockDim.x) {
        int idx = (kk < nk) ? keeplist[kk] : (NCAND - 1); // jax pad -> clamp N-1
        float4 bx = boxes[idx];
        float* o = outProps + ((size_t)b * POSTN + kk) * 4;
        o[0] = bx.x; o[1] = bx.y; o[2] = bx.z; o[3] = bx.w;
    }
}

// anchor<->gt matching: one wave per 16-anchor tile.
// inter computed in WMMA C layout; union = areaA[i] + areaG[j] - inter via
// v_wmma_f32_16x16x4_f32 (rank-2 outer sum through the matrix core).
// IoU ratio via v_rcp_f32 (branchless), then width-16 shuffle argmax.
static constexpr int TILES = A / 16;   // 4032 per image
__global__ void __launch_bounds__(256)
match_kernel(const float* __restrict__ gt, float* __restrict__ matchedOut) {
    int gid  = blockIdx.x * blockDim.x + threadIdx.x;
    int wave = gid >> 5;
    int lane = gid & 31;
    int b    = wave / TILES;
    int tile = wave - b * TILES;
    int base = tile * 16;
    int half = lane >> 4;      // 0: rows 0..7 / K0,K1   1: rows 8..15 / K2,K3
    int col  = lane & 15;

    // per-lane anchor rows for this half (C layout rows v + 8*half)
    float ax0[8], ay0[8], ax1[8], ay1[8];
#pragma unroll
    for (int v = 0; v < 8; v++)
        anchor_box(base + v + 8 * half, ax0[v], ay0[v], ax1[v], ay1[v]);

    // A matrix (16x4): row = col lane id; [areaA, 1, 0, 0]
    float areaRow;
    {
        float x0, y0, x1, y1;
        anchor_box(base + col, x0, y0, x1, y1);
        areaRow = (x1 - x0) * (y1 - y0);
    }
    v2f Am;
    Am.x = half ? 0.0f : areaRow;  // K0 (half0) / K2 (half1, zero)
    Am.y = half ? 0.0f : 1.0f;     // K1 / K3

    float bestV[8]; int bestI[8];
#pragma unroll
    for (int v = 0; v < 8; v++) { bestV[v] = -1e30f; bestI[v] = 0; }

    const float* gtb = gt + (size_t)b * M_GT * 5;
#pragma unroll
    for (int g = 0; g < 3; g++) {
        int j = g * 16 + col;
        float gx0 = 0.f, gy0 = 0.f, gx1 = 0.f, gy1 = 0.f, gc = -1.f;
        if (j < M_GT) {
            gx0 = gtb[j * 5 + 0]; gy0 = gtb[j * 5 + 1];
            gx1 = gtb[j * 5 + 2]; gy1 = gtb[j * 5 + 3];
            gc  = gtb[j * 5 + 4];
        }
        bool valid = (j < M_GT) && (gc != -1.0f);
        float gArea = (gx1 - gx0) * (gy1 - gy0);

        // B matrix (4x16): col = lane&15; [1, areaG, 0, 0]^T
        v2f Bm;
        Bm.x = half ? 0.0f : 1.0f;
        Bm.y = half ? 0.0f : gArea;

        float inter[8];
        v8f C;
#pragma unroll
        for (int v = 0; v < 8; v++) {
            float iw = fmaxf(fminf(ax1[v], gx1) - fmaxf(ax0[v], gx0), 0.0f);
            float ih = fmaxf(fminf(ay1[v], gy1) - fmaxf(ay0[v], gy0), 0.0f);
            inter[v] = iw * ih;
            C[v] = -inter[v];
        }
        // union[i][j] = areaA_i + areaG_j - inter  (D = A x B + C)
        v8f U = __builtin_amdgcn_wmma_f32_16x16x4_f32(
            false, Am, false, Bm, (short)0, C, false, false);
#pragma unroll
        for (int v = 0; v < 8; v++) {
            // branchless fast ratio: v_rcp_f32 + mul, then mask invalid cols
            float iou = inter[v] * __builtin_amdgcn_rcpf(U[v]);
            iou = valid ? iou : -1.0f;
            if (iou > bestV[v]) { bestV[v] = iou; bestI[v] = j; }  // strict > : first-max
        }
    }

    // argmax across the 16 lanes holding the same row (width-16 xor reduce)
#pragma unroll
    for (int v = 0; v < 8; v++) {
#pragma unroll
        for (int off = 8; off >= 1; off >>= 1) {
            float ov = __shfl_xor(bestV[v], off, 16);
            int   oi = __shfl_xor(bestI[v], off, 16);
            if (ov > bestV[v] || (ov == bestV[v] && oi < bestI[v])) {
                bestV[v] = ov; bestI[v] = oi;
            }
        }
    }

    // one lane per (half, v) writes the matched row
#pragma unroll
    for (int v = 0; v < 8; v++) {
        if (col == v) {
            int ai = base + v + 8 * half;
            float q = bestV[v]; int mi = bestI[v];
            float* o = matchedOut + ((size_t)b * A + ai) * 5;
            if (q <= 0.3f) {
                for (int c = 0; c < 5; c++) o[c] = -1.0f;
            } else if (q < 0.6f) {
                for (int c = 0; c < 5; c++) o[c] = -100000000.0f;
            } else {
                const float* gr = gtb + mi * 5;
                for (int c = 0; c < 5; c++) o[c] = gr[c];
            }
        }
    }
}

// ---------------- host launcher ----------------
extern "C" void kernel_launch(void* const* d_in, const int* in_sizes, int n_in,
                              void* d_out, int out_size, void* d_ws, size_t ws_size,
                              hipStream_t stream) {
    const float* d3 = (const float*)d_in[0];
    const float* s3 = (const float*)d_in[1];
    const float* d4 = (const float*)d_in[2];
    const float* s4 = (const float*)d_in[3];
    const float* d5 = (const float*)d_in[4];
    const float* s5 = (const float*)d_in[5];
    const float* gt = (const float*)d_in[6];
    float* out = (float*)d_out;

    char* ws = (char*)d_ws;
    size_t off = 0;
    auto alloc = [&](size_t bytes) -> char* {
        char* p = ws + off;
        off = (off + bytes + 255) & ~(size_t)255;
        return p;
    };
    float4* dec      = (float4*)alloc((size_t)B * A * sizeof(float4));
    ull*    packP3   = (ull*)   alloc((size_t)B * P3 * sizeof(ull));
    ull*    packP4   = (ull*)   alloc((size_t)B * P4 * sizeof(ull));
    ull*    packP5   = (ull*)   alloc((size_t)B * P5 * sizeof(ull));
    ull*    candPack = (ull*)   alloc((size_t)B * CPAD * sizeof(ull));
    float4* candBox  = (float4*)alloc((size_t)B * NCAND * sizeof(float4));
    float4* sortedBox= (float4*)alloc((size_t)B * NCAND * sizeof(float4));

    auto blocks = [](int total, int bs) { return (total + bs - 1) / bs; };

    // 1) decode + clip all anchors
    decode_kernel<<<blocks(B * A, 256), 256, 0, stream>>>(d3, d4, d5, dec);

    // 2) pack scores into sort keys (padded to power-of-two)
    fill_pack_kernel<<<blocks(B * P3, 256), 256, 0, stream>>>(s3, packP3, N3, P3);
    fill_pack_kernel<<<blocks(B * P4, 256), 256, 0, stream>>>(s4, packP4, N4, P4);
    fill_pack_kernel<<<blocks(B * P5, 256), 256, 0, stream>>>(s5, packP5, N5, P5);

    // bitonic sort (descending) over B independent power-of-two segments
    auto sort_segments = [&](ull* a, int P) {
        int total = B * P, segMask = P - 1;
        for (int k = 2; k <= P; k <<= 1) {
            int j = k >> 1;
            for (; j > 1024; j >>= 1)
                bitonic_global<<<blocks(total, 256), 256, 0, stream>>>(a, total, segMask, k, j);
            bitonic_fused<<<total / 2048, 1024, 0, stream>>>(a, segMask, k, j);
        }
    };
    sort_segments(packP3, P3);
    sort_segments(packP4, P4);
    sort_segments(packP5, P5);

    // 3) gather top-1000/level candidates, then sort the 3000 by score
    gather_cand<<<blocks(B * CPAD, 256), 256, 0, stream>>>(
        packP3, packP4, packP5, dec, candPack, candBox);
    sort_segments(candPack, CPAD);
    gather_sorted<<<blocks(B * NCAND, 256), 256, 0, stream>>>(candPack, candBox, sortedBox);

    // 4) NMS -> proposals (first out region)
    nms_kernel<<<B, 1024, 0, stream>>>(sortedBox, out);

    // 5) anchor<->gt matching via WMMA -> matched (second out region)
    float* matchedOut = out + (size_t)B * POSTN * 4;
    match_kernel<<<(B * TILES) / 8, 256, 0, stream>>>(gt, matchedOut);
}